// SGC_25709674233917
// MI455X (gfx1250) — compile-verified
//
#include <hip/hip_runtime.h>
#include <hip/hip_bf16.h>

// ---------------------------------------------------------------------------
// SGC-style fused graph conv for MI455X (gfx1250, wave32).
//
//   out = (sum_k alpha_k * A_k) @ x @ W + b
// rewritten (linear ops commute) as:
//   y   = x @ W                      (WMMA f32 16x16x4, full precision)
//   out = b  (broadcast init)
//   out[row] += alpha_k * val * y[col]   per edge (wave/edge, lane/channel)
//
// F = 64, C = 32 are hardwired (reference constants).
// ---------------------------------------------------------------------------

typedef __attribute__((ext_vector_type(2))) float v2f;
typedef __attribute__((ext_vector_type(8))) float v8f;

#define F_DIM 64
#define C_DIM 32

// ---------------------------------------------------------------------------
// Kernel A: y[N,32] = x[N,64] @ W[64,32] via V_WMMA_F32_16X16X4_F32.
// One wave computes one 16x16 output tile; 16 chained WMMAs over K=64.
// ---------------------------------------------------------------------------
__global__ __launch_bounds__(256) void sgc_xw_wmma(
    const float* __restrict__ x, const float* __restrict__ W,
    float* __restrict__ y, int N, int nTiles) {
  const int wid  = (int)((blockIdx.x * blockDim.x + threadIdx.x) >> 5);
  if (wid >= nTiles) return;                  // wave-uniform: EXEC stays full
  const int lane = (int)(threadIdx.x & 31);
  const int tn   = wid & 1;                   // N-tile (0..1), 16 cols each
  const int tm   = wid >> 1;                  // M-tile (0..N/16-1)

  const int l16  = lane & 15;
  const int h    = lane >> 4;                 // lane half (0/1)
  const int row  = tm * 16 + l16;             // A-matrix row for this lane
  const int rowc = row < N ? row : N - 1;     // clamp (keeps EXEC uniform)
  const int ncol = tn * 16 + l16;             // B/C column for this lane

  const float* xrow = x + (size_t)rowc * F_DIM;

  v8f acc = {};
#pragma unroll
  for (int kk = 0; kk < F_DIM; kk += 4) {
    const int k0 = kk + 2 * h;                // A: K = vgpr + 2*half
    v2f a = *(const v2f*)(xrow + k0);         // 8B-aligned (k0 even)
    v2f b;                                    // B: K = vgpr + 2*half, N = lane%16
    b.x = W[(size_t)k0 * C_DIM + ncol];
    b.y = W[(size_t)(k0 + 1) * C_DIM + ncol];
    acc = __builtin_amdgcn_wmma_f32_16x16x4_f32(
        /*neg_a=*/false, a, /*neg_b=*/false, b,
        /*c_mod=*/(short)0, acc, /*reuse_a=*/false, /*reuse_b=*/false);
  }

  // C/D layout: VGPR v -> M = v + 8*half, N = lane%16
#pragma unroll
  for (int v = 0; v < 8; ++v) {
    const int mOut = tm * 16 + v + 8 * h;
    if (mOut < N) y[(size_t)mOut * C_DIM + ncol] = acc[v];
  }
}

// ---------------------------------------------------------------------------
// Kernel B: out[n*32 + c] = b[c]  (fold the bias into the accumulator init)
// ---------------------------------------------------------------------------
__global__ __launch_bounds__(256) void sgc_bias_init(
    const float* __restrict__ b, float* __restrict__ out, int total) {
  const int i = (int)(blockIdx.x * blockDim.x + threadIdx.x);
  if (i < total) out[i] = b[i & (C_DIM - 1)];
}

// ---------------------------------------------------------------------------
// Kernel C: per edge e (flat over K*E):
//   out[rows[e]*32 + lane] += alpha[e/E] * vals[e] * y[cols[e]*32 + lane]
// One wave32 per edge, one lane per output channel: one coalesced 128B
// gather (L2-resident y) + 32 consecutive f32 atomics into one L2 line.
// ---------------------------------------------------------------------------
__global__ __launch_bounds__(256) void sgc_edge_agg(
    const float* __restrict__ vals, const int* __restrict__ rows,
    const int* __restrict__ cols, const float* __restrict__ alpha,
    const float* __restrict__ y, float* __restrict__ out,
    unsigned E, unsigned totalE) {
  const long long t = (long long)blockIdx.x * blockDim.x + threadIdx.x;
  const unsigned wid = (unsigned)(t >> 5);
  if (wid >= totalE) return;
  const int lane = (int)(t & 31);

  const unsigned k = wid / E;                 // hop index
  const float av  = alpha[k] * vals[wid];
  const int r = rows[wid];
  const int c = cols[wid];

  const float m = av * y[(size_t)c * C_DIM + lane];
  atomicAdd(&out[(size_t)r * C_DIM + lane], m);   // no-return global atomic
}

// ---------------------------------------------------------------------------
extern "C" void kernel_launch(void* const* d_in, const int* in_sizes, int n_in,
                              void* d_out, int out_size, void* d_ws, size_t ws_size,
                              hipStream_t stream) {
  const float* x     = (const float*)d_in[0];   // [N, 64]
  const float* vals  = (const float*)d_in[1];   // [K, E]
  const float* W     = (const float*)d_in[2];   // [64, 32]
  const float* bias  = (const float*)d_in[3];   // [32]
  const float* alpha = (const float*)d_in[4];   // [K]
  const int*   rows  = (const int*)d_in[5];     // [K, E]
  const int*   cols  = (const int*)d_in[6];     // [K, E]
  float* out = (float*)d_out;                   // [N, 32]
  float* y   = (float*)d_ws;                    // [N, 32] scratch (N*32*4 B)

  const int N       = in_sizes[0] / F_DIM;
  const int K       = in_sizes[4];
  const unsigned tE = (unsigned)in_sizes[1];    // K*E flat edges
  const unsigned E  = tE / (unsigned)(K > 0 ? K : 1);

  // Kernel A: WMMA projection y = x @ W
  const int mTiles = (N + 15) / 16;
  const int nTiles = mTiles * 2;                // C=32 -> two 16-wide N tiles
  {
    const long long threads = (long long)nTiles * 32;
    const int blocks = (int)((threads + 255) / 256);
    sgc_xw_wmma<<<blocks, 256, 0, stream>>>(x, W, y, N, nTiles);
  }

  // Kernel B: out = broadcast(bias)
  {
    const int total  = N * C_DIM;
    const int blocks = (total + 255) / 256;
    sgc_bias_init<<<blocks, 256, 0, stream>>>(bias, out, total);
  }

  // Kernel C: edge aggregation (wave per edge, lane per channel)
  {
    const long long threads = (long long)tE * 32;
    const int blocks = (int)((threads + 255) / 256);
    sgc_edge_agg<<<blocks, 256, 0, stream>>>(vals, rows, cols, alpha, y, out,
                                             E, tE);
  }
}